// SparseNodeLinear_50319836840312
// MI455X (gfx1250) — compile-verified
//
#include <hip/hip_runtime.h>

// Problem constants (fixed by the reference).
#define NN  2048   // num nodes
#define KK  2049   // N+1 inner dim
#define FF  32     // out features
#define BB  8      // batch
#define CAP 128    // max surviving indices per node (mean ~34; >>5-sigma safe)

typedef __attribute__((ext_vector_type(2))) float v2f;
typedef __attribute__((ext_vector_type(8))) float v8f;

#define WMMA_F32_4(A, Bm, Cm) \
    __builtin_amdgcn_wmma_f32_16x16x4_f32(false, (A), false, (Bm), (short)0, (Cm), false, false)

// Workspace layout: [ float zero[4] | int counts[NN] | int idx[NN][CAP] ]

// ---------------------------------------------------------------------------
// Pass 1: per-node mask-row compaction. One wave32 per node.
// Pads the index list to a multiple of 8 with index 0 (a valid gather address)
// and re-initializes the shared zero word (ws is poisoned by the harness).
// ---------------------------------------------------------------------------
__global__ __launch_bounds__(256)
void build_index_kernel(const unsigned char* __restrict__ mask,
                        float* __restrict__ zerow,
                        int* __restrict__ counts,
                        int* __restrict__ idx) {
    if (blockIdx.x == 0 && threadIdx.x < 4) zerow[threadIdx.x] = 0.0f;

    const int wave = threadIdx.x >> 5;
    const int lane = threadIdx.x & 31;
    const int node = blockIdx.x * 8 + wave;        // grid covers exactly NN nodes

    const unsigned char* row = mask + (size_t)node * KK;
    int* outIdx = idx + (size_t)node * CAP;

    int cnt = 0;
    for (int base = 0; base < KK; base += 32) {
        const int k = base + lane;
        const bool keep = (k < KK) && (row[k < KK ? k : 0] == 0);
        const unsigned m32 = __builtin_amdgcn_ballot_w32(keep);
        const int prefix = __popc(m32 & ((1u << lane) - 1u));
        if (keep) {
            const int slot = cnt + prefix;
            if (slot < CAP) outIdx[slot] = k;
        }
        cnt += __popc(m32);
    }
    cnt = (cnt < CAP) ? cnt : CAP;
    const int padded = (cnt + 7) & ~7;
    const int npad = ((padded <= CAP) ? padded : CAP) - cnt;
    if (lane < npad) outIdx[cnt + lane] = 0;       // safe dummy index
    if (lane == 0) counts[node] = cnt;
}

// ---------------------------------------------------------------------------
// Pass 2: gathered per-node GEMM on V_WMMA_F32_16X16X4_F32. One wave per node.
//   A (16x4 f32): M = batch. Rows 8..15 gather from a ws zero word via a
//     masked offset -> they load 0.0f with no predicate in the dataflow.
//   B (4x16 f32): same per-lane k-slot mapping; N = feature tile (f0=0,16).
//   C/D (16x16 f32): VGPR r, lanes 0-15 -> (M=r, N=lane): exactly batches 0..7.
// Main loop is fully branch- and select-free; the <=7-slot tail runs once
// under a wave-uniform guard (EXEC stays full for all WMMAs).
// ---------------------------------------------------------------------------
__global__ __launch_bounds__(256)
void sparse_node_wmma_kernel(const float* __restrict__ x,
                             const float* __restrict__ weight,
                             const float* __restrict__ bias,
                             const float* __restrict__ zerow,
                             const int* __restrict__ counts,
                             const int* __restrict__ idx,
                             float* __restrict__ out) {
    const int wave   = threadIdx.x >> 5;
    const int lane   = threadIdx.x & 31;
    const int node   = blockIdx.x * 8 + wave;      // grid covers exactly NN nodes
    const int half   = lane >> 4;                  // 0: k-slots {0,1}, 1: {2,3}
    const int laneLo = lane & 15;

    const int  cnt = counts[node];                 // wave-uniform
    const int* row = idx + (size_t)node * CAP;
    const float* w0 = weight + (size_t)node * (KK * FF) + laneLo;  // f in [0,16)
    const float* w1 = w0 + 16;                                     // f in [16,32)

    const bool aval = (laneLo < BB);               // batch rows 8..15 are padding
    int amask = aval ? ~0 : 0;                     // zeroes gather offset below
    asm("" : "+v"(amask));                         // opaque: keep it a v_and, not a branch
    const float* xb = aval ? (x + (size_t)laneLo * ((size_t)KK * NN) + node)
                           : zerow;                // padded rows read 0.0f

    v8f c0 = {};
    v8f c1 = {};

    // ---- main loop: all 8 slots valid, no predicates anywhere ----
    const int nfull = cnt >> 3;
    for (int it = 0; it < nfull; ++it) {
        const int j0 = (it << 3) + (half << 1);
        const int2 ka = *(const int2*)(row + j0);      // 8B-aligned pair load
        const int2 kb = *(const int2*)(row + j0 + 4);
        {
            v2f a, b0, b1;
            a.x  = xb[(ka.x * NN) & amask];
            a.y  = xb[(ka.y * NN) & amask];
            b0.x = w0[ka.x * FF];
            b0.y = w0[ka.y * FF];
            b1.x = w1[ka.x * FF];
            b1.y = w1[ka.y * FF];
            c0 = WMMA_F32_4(a, b0, c0);
            c1 = WMMA_F32_4(a, b1, c1);
        }
        {
            v2f a, b0, b1;
            a.x  = xb[(kb.x * NN) & amask];
            a.y  = xb[(kb.y * NN) & amask];
            b0.x = w0[kb.x * FF];
            b0.y = w0[kb.y * FF];
            b1.x = w1[kb.x * FF];
            b1.y = w1[kb.y * FF];
            c0 = WMMA_F32_4(a, b0, c0);
            c1 = WMMA_F32_4(a, b1, c1);
        }
    }

    // ---- tail: up to 7 slots; index list padded to a multiple of 8 ----
    if (cnt & 7) {                                  // wave-uniform branch
        const int j0 = (nfull << 3) + (half << 1);
        const int j1 = j0 + 4;
        const int2 ka = *(const int2*)(row + j0);
        const int2 kb = *(const int2*)(row + j1);
        {
            const float va0 = xb[(ka.x * NN) & amask];
            const float va1 = xb[(ka.y * NN) & amask];
            v2f a, b0, b1;
            a.x  = ((j0 + 0) < cnt) ? va0 : 0.0f;
            a.y  = ((j0 + 1) < cnt) ? va1 : 0.0f;
            b0.x = w0[ka.x * FF];
            b0.y = w0[ka.y * FF];
            b1.x = w1[ka.x * FF];
            b1.y = w1[ka.y * FF];
            c0 = WMMA_F32_4(a, b0, c0);
            c1 = WMMA_F32_4(a, b1, c1);
        }
        {
            const float va0 = xb[(kb.x * NN) & amask];
            const float va1 = xb[(kb.y * NN) & amask];
            v2f a, b0, b1;
            a.x  = ((j1 + 0) < cnt) ? va0 : 0.0f;
            a.y  = ((j1 + 1) < cnt) ? va1 : 0.0f;
            b0.x = w0[kb.x * FF];
            b0.y = w0[kb.y * FF];
            b1.x = w1[kb.x * FF];
            b1.y = w1[kb.y * FF];
            c0 = WMMA_F32_4(a, b0, c0);
            c1 = WMMA_F32_4(a, b1, c1);
        }
    }

    // Epilogue: rows r=0..7 in lanes 0..15 are batches 0..7; add bias, store.
    if (half == 0) {
        const float bv0 = bias[(size_t)laneLo * NN + node];
        const float bv1 = bias[(size_t)(16 + laneLo) * NN + node];
#pragma unroll
        for (int r = 0; r < 8; ++r) {
            out[(size_t)r * (FF * NN) + (size_t)laneLo * NN + node]        = c0[r] + bv0;
            out[(size_t)r * (FF * NN) + (size_t)(16 + laneLo) * NN + node] = c1[r] + bv1;
        }
    }
}

// ---------------------------------------------------------------------------
extern "C" void kernel_launch(void* const* d_in, const int* in_sizes, int n_in,
                              void* d_out, int out_size, void* d_ws, size_t ws_size,
                              hipStream_t stream) {
    const float*         x      = (const float*)d_in[0];         // [B, K, N]
    const float*         weight = (const float*)d_in[1];         // [N, K, F]
    const float*         bias   = (const float*)d_in[2];         // [F, N]
    const unsigned char* mask   = (const unsigned char*)d_in[3]; // [N, K] bool
    float*               out    = (float*)d_out;                 // [B, F, N]

    // Workspace layout: zero word (16B) | counts[NN] | idx[NN][CAP] (~1.06 MB).
    float* zerow  = (float*)d_ws;
    int*   counts = (int*)d_ws + 4;
    int*   idx    = counts + NN;

    build_index_kernel<<<NN / 8, 256, 0, stream>>>(mask, zerow, counts, idx);
    sparse_node_wmma_kernel<<<NN / 8, 256, 0, stream>>>(x, weight, bias, zerow,
                                                        counts, idx, out);
}